// S2VTModel_41077067219320
// MI455X (gfx1250) — compile-verified
//
#include <hip/hip_runtime.h>
#include <hip/hip_bf16.h>
#include <math.h>

// Problem constants (fixed by the reference)
#define B_   64
#define T_   32
#define DV_  2048
#define H_   1024
#define DW_  512
#define V_   20000
#define ML_  3
#define G3H  (3 * H_)

typedef __attribute__((ext_vector_type(2))) float v2f;
typedef __attribute__((ext_vector_type(8))) float v8f;

#define WMMA_F32(a, b, c) \
    __builtin_amdgcn_wmma_f32_16x16x4_f32(false, (a), false, (b), (short)0, (c), false, false)

// ---------------------------------------------------------------------------
// fp32 WMMA GEMM, register-blocked 2x2:  C[M,N] = A[M,K] @ W[N,K]^T + bias[N]
// One wave (32 threads) per 32x32 output tile. grid = (N/32, M/32).
// Per k-step: 4 b64 loads feed 4 WMMAs -> 1 VMEM per WMMA (8 FLOP/byte).
// Fragment layout (V_WMMA_F32_16X16X4_F32):
//   A(16x4):  lane l holds (m = l&15, k = 2*(l>>4) + {0,1})  -> contiguous f2
//   B(4x16):  lane l holds (n = l&15, k = 2*(l>>4) + {0,1})  -> contiguous f2 of W row n
//   C/D:      vgpr r, lane l -> (m = r + 8*(l>>4), n = l&15)
// ---------------------------------------------------------------------------
__global__ __launch_bounds__(32)
void gemm_wmma_f32(const float* __restrict__ A, int lda,
                   const float* __restrict__ W, int ldw,
                   const float* __restrict__ bias,
                   float* __restrict__ C, int ldc, int K)
{
    const int lane = threadIdx.x;
    const int hi = lane >> 4;
    const int lo = lane & 15;
    const int col0 = blockIdx.x * 32;
    const int row0 = blockIdx.y * 32;

    const float* a0p = A + (size_t)(row0 + lo) * lda + 2 * hi;
    const float* a1p = a0p + (size_t)16 * lda;
    const float* b0p = W + (size_t)(col0 + lo) * ldw + 2 * hi;
    const float* b1p = b0p + (size_t)16 * ldw;

    v8f c00 = {}, c01 = {}, c10 = {}, c11 = {};
#pragma unroll 2
    for (int k = 0; k < K; k += 4) {
        v2f a0 = *reinterpret_cast<const v2f*>(a0p + k);
        v2f a1 = *reinterpret_cast<const v2f*>(a1p + k);
        v2f b0 = *reinterpret_cast<const v2f*>(b0p + k);
        v2f b1 = *reinterpret_cast<const v2f*>(b1p + k);
        c00 = WMMA_F32(a0, b0, c00);
        c01 = WMMA_F32(a0, b1, c01);
        c10 = WMMA_F32(a1, b0, c10);
        c11 = WMMA_F32(a1, b1, c11);
    }

    const float bv0 = bias[col0 + lo];
    const float bv1 = bias[col0 + 16 + lo];
#pragma unroll
    for (int r = 0; r < 8; ++r) {
        const int m0 = row0 + r + 8 * hi;
        const int m1 = m0 + 16;
        C[(size_t)m0 * ldc + col0 + lo]      = c00[r] + bv0;
        C[(size_t)m0 * ldc + col0 + 16 + lo] = c01[r] + bv1;
        C[(size_t)m1 * ldc + col0 + lo]      = c10[r] + bv0;
        C[(size_t)m1 * ldc + col0 + 16 + lo] = c11[r] + bv1;
    }
}

// ---------------------------------------------------------------------------
// Fused GRU step, 32 batch rows x 16 cols per wave (2 row tiles x 3 gates =
// 6 WMMA accumulators fed by 5 loads per k-step):
//   hp = h_prev @ W_hh^T + b_hh, then
//   r=sig(xr+hr), z=sig(xz+hz), n=tanh(xn+r*hn), h' = (1-z)*n + z*h_prev
// xp is pre-projected input (already includes b_ih); xps = row stride
// (xps==0 broadcasts one row -> decoder's b_ih1-only input).
// grid = (H/16, B/32), 32 threads.
// ---------------------------------------------------------------------------
__global__ __launch_bounds__(32)
void gru_step_wmma(const float* __restrict__ xp, int xps,
                   const float* __restrict__ h_prev,
                   const float* __restrict__ W_hh,   // (3H, H) row-major
                   const float* __restrict__ b_hh,   // (3H)
                   float* __restrict__ h_next)       // (B, H)
{
    const int lane = threadIdx.x;
    const int hi = lane >> 4;
    const int lo = lane & 15;
    const int col0 = blockIdx.x * 16;   // within H
    const int row0 = blockIdx.y * 32;   // batch rows

    const float* a0p = h_prev + (size_t)(row0 + lo) * H_ + 2 * hi;
    const float* a1p = a0p + (size_t)16 * H_;
    const float* wr  = W_hh + (size_t)(col0 + lo) * H_ + 2 * hi;
    const float* wz  = wr + (size_t)H_ * H_;
    const float* wn  = wz + (size_t)H_ * H_;

    v8f r0 = {}, z0 = {}, n0 = {};
    v8f r1 = {}, z1 = {}, n1 = {};
#pragma unroll 2
    for (int k = 0; k < H_; k += 4) {
        v2f a0 = *reinterpret_cast<const v2f*>(a0p + k);
        v2f a1 = *reinterpret_cast<const v2f*>(a1p + k);
        v2f br = *reinterpret_cast<const v2f*>(wr + k);
        v2f bz = *reinterpret_cast<const v2f*>(wz + k);
        v2f bn = *reinterpret_cast<const v2f*>(wn + k);
        r0 = WMMA_F32(a0, br, r0);
        z0 = WMMA_F32(a0, bz, z0);
        n0 = WMMA_F32(a0, bn, n0);
        r1 = WMMA_F32(a1, br, r1);
        z1 = WMMA_F32(a1, bz, z1);
        n1 = WMMA_F32(a1, bn, n1);
    }

    const int col = col0 + lo;
    const float bhr = b_hh[col];
    const float bhz = b_hh[H_ + col];
    const float bhn = b_hh[2 * H_ + col];

#pragma unroll
    for (int g = 0; g < 2; ++g) {
        const v8f& aR = g ? r1 : r0;
        const v8f& aZ = g ? z1 : z0;
        const v8f& aN = g ? n1 : n0;
#pragma unroll
        for (int r = 0; r < 8; ++r) {
            const int row = row0 + 16 * g + r + 8 * hi;
            const float* xrow = xp + (size_t)row * xps;
            const float hr = aR[r] + bhr;
            const float hz = aZ[r] + bhz;
            const float hn = aN[r] + bhn;
            const float rg = 1.0f / (1.0f + expf(-(xrow[col] + hr)));
            const float zg = 1.0f / (1.0f + expf(-(xrow[H_ + col] + hz)));
            const float ng = tanhf(xrow[2 * H_ + col] + rg * hn);
            const float hp = h_prev[(size_t)row * H_ + col];
            h_next[(size_t)row * H_ + col] = (1.0f - zg) * ng + zg * hp;
        }
    }
}

// ---------------------------------------------------------------------------
// Row-wise log-softmax over V_=20000; one block per batch row.
// out points at d_out + i*V; row b written at out + b*(ML_*V_).
// ---------------------------------------------------------------------------
__global__ __launch_bounds__(256)
void log_softmax_row(const float* __restrict__ logits, float* __restrict__ out)
{
    __shared__ float red[256];
    const int b = blockIdx.x;
    const float* x = logits + (size_t)b * V_;
    float m = -3.402823466e38f;
    for (int v = threadIdx.x; v < V_; v += 256) m = fmaxf(m, x[v]);
    red[threadIdx.x] = m; __syncthreads();
    for (int s = 128; s > 0; s >>= 1) {
        if (threadIdx.x < s) red[threadIdx.x] = fmaxf(red[threadIdx.x], red[threadIdx.x + s]);
        __syncthreads();
    }
    m = red[0]; __syncthreads();
    float sum = 0.0f;
    for (int v = threadIdx.x; v < V_; v += 256) sum += expf(x[v] - m);
    red[threadIdx.x] = sum; __syncthreads();
    for (int s = 128; s > 0; s >>= 1) {
        if (threadIdx.x < s) red[threadIdx.x] += red[threadIdx.x + s];
        __syncthreads();
    }
    const float lse = m + logf(red[0]);
    float* o = out + (size_t)b * (ML_ * V_);
    for (int v = threadIdx.x; v < V_; v += 256) o[v] = x[v] - lse;
}

// ---------------------------------------------------------------------------
extern "C" void kernel_launch(void* const* d_in, const int* in_sizes, int n_in,
                              void* d_out, int out_size, void* d_ws, size_t ws_size,
                              hipStream_t stream)
{
    const float* vid   = (const float*)d_in[0];
    const float* W_ih1 = (const float*)d_in[1];
    const float* W_hh1 = (const float*)d_in[2];
    const float* b_ih1 = (const float*)d_in[3];
    const float* b_hh1 = (const float*)d_in[4];
    const float* W_ih2 = (const float*)d_in[5];   // (3H, H+DW); only [:, :H] used
    const float* W_hh2 = (const float*)d_in[6];
    const float* b_ih2 = (const float*)d_in[7];
    const float* b_hh2 = (const float*)d_in[8];
    const float* W_obj = (const float*)d_in[9];
    const float* b_obj = (const float*)d_in[10];
    const float* W_rel = (const float*)d_in[11];
    const float* b_rel = (const float*)d_in[12];
    float* out = (float*)d_out;

    // Workspace carve-up (~66 MB of fp32)
    float* ws     = (float*)d_ws;
    float* xp1    = ws;                                  // (B,T,3H)
    float* out1   = xp1  + (size_t)B_ * T_ * G3H;        // (T,B,H)  (doubles as h1 history)
    float* xp2    = out1 + (size_t)T_ * B_ * H_;         // (T,B,3H)
    float* zeroh  = xp2  + (size_t)T_ * B_ * G3H;        // (B,H) zeros
    float* h2a    = zeroh + (size_t)B_ * H_;
    float* h2b    = h2a   + (size_t)B_ * H_;
    float* h1a    = h2b   + (size_t)B_ * H_;
    float* h1b    = h1a   + (size_t)B_ * H_;
    float* xp2t   = h1b   + (size_t)B_ * H_;             // (B,3H)
    float* logits = xp2t  + (size_t)B_ * G3H;            // (B,V)

    hipMemsetAsync(zeroh, 0, (size_t)B_ * H_ * sizeof(float), stream);

    // Stage A: xp1 = vid @ W_ih1^T + b_ih1 ; rows r = b*T + t (natural (B,T,3H))
    gemm_wmma_f32<<<dim3(G3H / 32, (B_ * T_) / 32), 32, 0, stream>>>(
        vid, DV_, W_ih1, DV_, b_ih1, xp1, G3H, DV_);

    // Layer-1 scan: h_next[t] stored as out1[t]
    for (int t = 0; t < T_; ++t) {
        const float* hp = (t == 0) ? zeroh : out1 + (size_t)(t - 1) * B_ * H_;
        gru_step_wmma<<<dim3(H_ / 16, B_ / 32), 32, 0, stream>>>(
            xp1 + (size_t)t * G3H, T_ * G3H, hp, W_hh1, b_hh1,
            out1 + (size_t)t * B_ * H_);
    }

    // Stage C: xp2 = out1 @ W_ih2[:, :H]^T + b_ih2 ; rows r = t*B + b
    gemm_wmma_f32<<<dim3(G3H / 32, (T_ * B_) / 32), 32, 0, stream>>>(
        out1, H_, W_ih2, H_ + DW_, b_ih2, xp2, G3H, H_);

    // Layer-2 scan (ping-pong state)
    float* h2prev = zeroh;
    float* h2next = h2a;
    for (int t = 0; t < T_; ++t) {
        gru_step_wmma<<<dim3(H_ / 16, B_ / 32), 32, 0, stream>>>(
            xp2 + (size_t)t * B_ * G3H, G3H, h2prev, W_hh2, b_hh2, h2next);
        h2prev = h2next;
        h2next = (h2next == h2a) ? h2b : h2a;
    }

    // Decoder (MAXLEN steps)
    float* h1prev = out1 + (size_t)(T_ - 1) * B_ * H_;  // final layer-1 state
    float* h1next = h1a;
    for (int i = 0; i < ML_; ++i) {
        // h1 = GRU1(broadcast b_ih1, h1)
        gru_step_wmma<<<dim3(H_ / 16, B_ / 32), 32, 0, stream>>>(
            b_ih1, 0, h1prev, W_hh1, b_hh1, h1next);
        // xp2_t = h1 @ W_ih2[:, :H]^T + b_ih2
        gemm_wmma_f32<<<dim3(G3H / 32, B_ / 32), 32, 0, stream>>>(
            h1next, H_, W_ih2, H_ + DW_, b_ih2, xp2t, G3H, H_);
        // h2 = GRU2(xp2_t, h2)
        gru_step_wmma<<<dim3(H_ / 16, B_ / 32), 32, 0, stream>>>(
            xp2t, G3H, h2prev, W_hh2, b_hh2, h2next);
        // logits = h2 @ W^T + b   (V = 20000 = 32 * 625)
        const float* Wl = (i == 1) ? W_rel : W_obj;
        const float* bl = (i == 1) ? b_rel : b_obj;
        gemm_wmma_f32<<<dim3(V_ / 32, B_ / 32), 32, 0, stream>>>(
            h2next, H_, Wl, H_, bl, logits, V_, H_);
        // out[:, i, :] = log_softmax(logits)
        log_softmax_row<<<B_, 256, 0, stream>>>(logits, out + (size_t)i * V_);

        h1prev = h1next;
        h1next = (h1next == h1a) ? h1b : h1a;
        float* tmp = h2prev; h2prev = h2next; h2next = tmp;
    }
}